// CGCoupler_55508157333853
// MI455X (gfx1250) — compile-verified
//
#include <hip/hip_runtime.h>
#include <stdint.h>

// ---------------------------------------------------------------------------
// CG tensor-product coupler for gfx1250 (MI455X).
//   out[n, ro[m]] += cg[m] * x1[n, r1[m]] * x2[n, r2[m]]
// Memory-bound (~50 FLOP/KB) -> no WMMA. Uses CDNA5 async global<->LDS DMA
// (b128, ASYNCcnt), LDS scatter-add (ds_add_f32), wave32 tiling.
// ---------------------------------------------------------------------------

#define ROWS     8            // batch rows per workgroup tile (1 wave stages 1 row)
#define REP_DIM  640          // fixed by reference METADATA [64,64,32,32]
#define STRIDE   644          // padded LDS row stride (floats): 16B-aligned, 4-bank skew
#define BLOCK    256          // 8 wave32 per block
#define NCHUNK   32           // index-list chunks per block
#define X2_OFF   (ROWS * STRIDE)          // 5152 floats
#define OUT_OFF  (2 * ROWS * STRIDE)      // 10304 floats
#define SMEM_WORDS (3 * ROWS * STRIDE)    // 15456 floats = 61824 B

// -------- pack (cg, r1*4, r2*4, ro*4) into one 16B AoS entry in d_ws -------
__global__ __launch_bounds__(BLOCK) void cg_pack(
    const float* __restrict__ cg, const int* __restrict__ r1,
    const int* __restrict__ r2, const int* __restrict__ ro,
    uint4* __restrict__ tab, int M) {
  int m = blockIdx.x * BLOCK + threadIdx.x;
  if (m < M) {
    uint4 e;
    e.x = __float_as_uint(cg[m]);
    e.y = (unsigned)r1[m] << 2;   // byte offsets: saves shifts in the hot loop
    e.z = (unsigned)r2[m] << 2;
    e.w = (unsigned)ro[m] << 2;
    tab[m] = e;
  }
}

// Low 32 bits of a flat LDS pointer == workgroup-relative LDS byte offset
// (ISA 10.2: the LDS aperture uses the low 32 address bits).
__device__ __forceinline__ unsigned lds_off(const void* p) {
  return (unsigned)(unsigned long long)p;
}

// Async DMA global->LDS (per-lane LDS address, tracked by ASYNCcnt).
__device__ __forceinline__ void async_g2l_b128(const float* g, const float* l) {
  asm volatile("global_load_async_to_lds_b128 %0, %1, off"
               :: "v"(lds_off(l)), "v"(g) : "memory");
}
__device__ __forceinline__ void async_g2l_b32(const float* g, const float* l) {
  asm volatile("global_load_async_to_lds_b32 %0, %1, off"
               :: "v"(lds_off(l)), "v"(g) : "memory");
}
// Async DMA LDS->global.
__device__ __forceinline__ void async_l2g_b128(float* g, const float* l) {
  asm volatile("global_store_async_from_lds_b128 %0, %1, off"
               :: "v"(g), "v"(lds_off(l)) : "memory");
}
__device__ __forceinline__ void async_l2g_b32(float* g, const float* l) {
  asm volatile("global_store_async_from_lds_b32 %0, %1, off"
               :: "v"(g), "v"(lds_off(l)) : "memory");
}
__device__ __forceinline__ void wait_async0() {
#if __has_builtin(__builtin_amdgcn_s_wait_asynccnt)
  __builtin_amdgcn_s_wait_asynccnt(0);
#else
  asm volatile("s_wait_asynccnt 0x0" ::: "memory");
#endif
}

// -------- main coupling kernel ---------------------------------------------
__global__ __launch_bounds__(BLOCK) void cg_couple(
    const float* __restrict__ x1, const float* __restrict__ x2,
    const uint4* __restrict__ tab, float* __restrict__ out,
    int M, int N, int out_dim) {
  __shared__ float smem[SMEM_WORDS];

  const int tid  = threadIdx.x;
  const int lrow = tid >> 5;                 // staging: wave id == row in tile
  const int lc   = tid & 31;                 // staging: lane -> 16B chunk
  const long long rowg = (long long)blockIdx.x * ROWS + lrow;
  const bool valid = rowg < (long long)N;

  // Zero the accumulator tile with b128 DS stores (overlaps with DMA below).
  {
    float4 z; z.x = z.y = z.z = z.w = 0.0f;
    float4* zp = (float4*)&smem[OUT_OFF];
    for (int k = tid; k < ROWS * STRIDE / 4; k += BLOCK) zp[k] = z;
  }

  // Stage x1/x2 rows into LDS: one wave per row, 5 coalesced b128 DMAs
  // per input per lane-iteration (512B bursts per wave).
  if (valid) {
    const float* g1 = x1 + rowg * REP_DIM;
    const float* g2 = x2 + rowg * REP_DIM;
    float* l1 = &smem[lrow * STRIDE];
    float* l2 = &smem[X2_OFF + lrow * STRIDE];
#pragma unroll
    for (int k = 0; k < REP_DIM / 128; ++k) {  // 5 iterations
      int f = (lc + k * 32) * 4;               // float index, 16B-aligned
      async_g2l_b128(g1 + f, l1 + f);
      async_g2l_b128(g2 + f, l2 + f);
    }
  }
  wait_async0();
  __syncthreads();

  // Gather * cg -> scatter-add into LDS. Lane mapping: r = tid&7 (row),
  // c = tid>>3 (chunk). 8 lanes share each 16B table entry (one cacheline
  // per wave per iteration). Row skew (stride 644 -> 4 banks) keeps the
  // 8-lane gather conflict-free; ds_add_f32 resolves collisions.
  {
    const int r = tid & (ROWS - 1);
    const int c = tid >> 3;
    const char* s1 = (const char*)&smem[r * STRIDE];
    const char* s2 = (const char*)&smem[X2_OFF + r * STRIDE];
    char*       so = (char*)&smem[OUT_OFF + r * STRIDE];
    for (int m = c; m < M; m += NCHUNK) {
      uint4 e = tab[m];
      float v = __uint_as_float(e.x)
              * *(const float*)(s1 + e.y)
              * *(const float*)(s2 + e.z);
      __hip_atomic_fetch_add((float*)(so + e.w), v,
                             __ATOMIC_RELAXED, __HIP_MEMORY_SCOPE_WORKGROUP);
    }
  }
  __syncthreads();

  // Async write-back of the output tile (LDS -> global), b128 bursts.
  if (valid) {
    float* go = out + rowg * (long long)out_dim;
    const float* lo = &smem[OUT_OFF + lrow * STRIDE];
    if ((out_dim & 3) == 0) {
      for (int f = lc * 4; f < out_dim; f += 128)
        async_l2g_b128(go + f, lo + f);
    } else {
      for (int f = lc; f < out_dim; f += 32)
        async_l2g_b32(go + f, lo + f);
    }
  }
  wait_async0();
}

// ---------------------------------------------------------------------------
extern "C" void kernel_launch(void* const* d_in, const int* in_sizes, int n_in,
                              void* d_out, int out_size, void* d_ws, size_t ws_size,
                              hipStream_t stream) {
  const float* x1 = (const float*)d_in[0];
  const float* x2 = (const float*)d_in[1];
  const float* cg = (const float*)d_in[2];
  const int*   r1 = (const int*)d_in[3];
  const int*   r2 = (const int*)d_in[4];
  const int*   ro = (const int*)d_in[5];
  // d_in[6] is the out_dim scalar on device; derive dims on host instead.
  const int M = in_sizes[2];
  const int N = in_sizes[0] / REP_DIM;
  const int out_dim = out_size / N;

  uint4* tab = (uint4*)d_ws;  // M * 16 bytes of scratch

  cg_pack<<<(M + BLOCK - 1) / BLOCK, BLOCK, 0, stream>>>(cg, r1, r2, ro, tab, M);

  const int nblk = (N + ROWS - 1) / ROWS;
  cg_couple<<<nblk, BLOCK, 0, stream>>>(x1, x2, tab, (float*)d_out, M, N, out_dim);
}